// ResidualVQ_27693949125326
// MI455X (gfx1250) — compile-verified
//
#include <hip/hip_runtime.h>
#include <hip/hip_bf16.h>
#include <math.h>

typedef float v2f __attribute__((ext_vector_type(2)));
typedef float v8f __attribute__((ext_vector_type(8)));

#define NT_ROWS 32768   // N*T rows
#define CDIM    512     // channels (K of GEMM)
#define KCODES  1024    // codebook entries (N of GEMM)
#define QSTAGES 6
#define NB      64      // batch
#define TDIM    512     // time

// ---------------------------------------------------------------------------
// (N,C,T) -> row-major (N*T, C) residual, via LDS-tiled transpose
// ---------------------------------------------------------------------------
__global__ void vq_transpose_in(const float* __restrict__ x, float* __restrict__ residual) {
    __shared__ float tile[32][33];
    const int n = blockIdx.z;
    const int c0 = blockIdx.y * 32, t0 = blockIdx.x * 32;
    const int tx = threadIdx.x, ty = threadIdx.y;
    const float* xp = x + (size_t)n * CDIM * TDIM;
#pragma unroll
    for (int j = 0; j < 32; j += 8)
        tile[ty + j][tx] = xp[(size_t)(c0 + ty + j) * TDIM + (t0 + tx)];
    __syncthreads();
#pragma unroll
    for (int j = 0; j < 32; j += 8)
        residual[((size_t)n * TDIM + (t0 + ty + j)) * CDIM + (c0 + tx)] = tile[tx][ty + j];
}

// ---------------------------------------------------------------------------
// quantized_out[n][c][t] = x[n][c][t] - residual_final[(n*T+t)*C + c]
// ---------------------------------------------------------------------------
__global__ void vq_transpose_out(const float* __restrict__ x, const float* __restrict__ residual,
                                 float* __restrict__ outq) {
    __shared__ float tile[32][33];
    const int n = blockIdx.z;
    const int c0 = blockIdx.y * 32, t0 = blockIdx.x * 32;
    const int tx = threadIdx.x, ty = threadIdx.y;
#pragma unroll
    for (int j = 0; j < 32; j += 8)
        tile[ty + j][tx] = residual[((size_t)n * TDIM + (t0 + ty + j)) * CDIM + (c0 + tx)];
    __syncthreads();
    const float* xp = x + (size_t)n * CDIM * TDIM;
    float* op = outq + (size_t)n * CDIM * TDIM;
#pragma unroll
    for (int j = 0; j < 32; j += 8) {
        size_t e = (size_t)(c0 + ty + j) * TDIM + (t0 + tx);
        op[e] = xp[e] - tile[tx][ty + j];
    }
}

// ---------------------------------------------------------------------------
// Per-stage init: hn[k] = 0.5*||c_k||^2, zero counts + loss accumulator.
// grid = 128 blocks x 256 threads, one wave32 per code row.
// ---------------------------------------------------------------------------
__global__ void vq_stage_init(const float* __restrict__ cb, float* __restrict__ hn,
                              float* __restrict__ counts, float* __restrict__ loss_accum) {
    const int wave = threadIdx.x >> 5, lane = threadIdx.x & 31;
    const int code = blockIdx.x * 8 + wave;
    if (blockIdx.x < 4) counts[blockIdx.x * 256 + threadIdx.x] = 0.0f;
    if (blockIdx.x == 4 && threadIdx.x == 0) loss_accum[0] = 0.0f;
    const float4* c = (const float4*)(cb + (size_t)code * CDIM);
    float s = 0.f;
#pragma unroll
    for (int j = 0; j < 4; ++j) {
        float4 v = c[lane + 32 * j];
        s += v.x * v.x + v.y * v.y + v.z * v.z + v.w * v.w;
    }
#pragma unroll
    for (int m = 16; m >= 1; m >>= 1) s += __shfl_xor(s, m, 32);
    if (lane == 0) hn[code] = 0.5f * s;
}

// ---------------------------------------------------------------------------
// Fused f32 GEMM (R x CB^T) + running argmax of (r.c - 0.5||c||^2).
//
// WG = 128 threads (4 wave32). Each wave owns a 16-row M-tile in LDS in the
// natural WMMA A-fragment order: one k-quad = 64 contiguous floats = the 64
// LDS banks, so fragment ds_load_b64s are conflict-free AND every k-step is
// a constant immediate offset (kq*256B <= 32512) from one base register.
// Staging maps lanes across the m/n dimension -> conflict-free ds stores,
// with 128B-contiguous global segments per row. 4 accumulators break the
// WMMA RAW chain for 4-deep XDL pipelining.
// LDS: 4*32KB (A) + 32KB (B) = 160KB.
// ---------------------------------------------------------------------------
__launch_bounds__(128)
__global__ void vq_gemm_argmax(const float* __restrict__ R, const float* __restrict__ cb,
                               const float* __restrict__ hn,
                               float* __restrict__ best_score, int* __restrict__ best_idx) {
    __shared__ float ldsA[4 * 8192];
    __shared__ float ldsB[8192];
    const int t    = threadIdx.x;
    const int lane = t & 31;
    const int wave = t >> 5;
    const int m0   = blockIdx.x * 64;

    // ---- Stage A: 64 rows x 512 floats. lds[(row>>4)*8192 + kq*64 + (row&15)*4 + j]
    // lanes span rows (conflict-free stores); per row a group of 8 lanes reads
    // 128B contiguous global memory.
#pragma unroll 4
    for (int i = 0; i < 64; ++i) {
        int row = (t & 15) + 16 * (i & 3);
        int kq  = (t >> 4) + 8 * (i >> 2);
        float4 v = *(const float4*)(R + (size_t)(m0 + row) * CDIM + kq * 4);
        *(float4*)(ldsA + (row >> 4) * 8192 + kq * 64 + ((row & 15) << 2)) = v;
    }

    const int laneOff = ((lane & 15) << 2) + ((lane >> 4) << 1);
    const float* aBase = ldsA + wave * 8192 + laneOff;
    const float* bBase = ldsB + laneOff;

    float bval[8];
    int   bidx[8];
#pragma unroll
    for (int v = 0; v < 8; ++v) { bval[v] = -3.4e38f; bidx[v] = 0; }

    for (int nt = 0; nt < 64; ++nt) {
        __syncthreads();   // previous tile consumed (also covers A staging on nt==0)
        // ---- Stage B tile: 16 codes x 512 floats, same layout/mapping as A.
#pragma unroll
        for (int i = 0; i < 16; ++i) {
            int n  = t & 15;
            int kq = (t >> 4) + 8 * i;
            float4 v = *(const float4*)(cb + (size_t)(nt * 16 + n) * CDIM + kq * 4);
            *(float4*)(ldsB + kq * 64 + (n << 2)) = v;
        }
        __syncthreads();
        if (nt + 1 < 64)
            __builtin_prefetch(cb + (size_t)((nt + 1) * 16) * CDIM + lane * 16, 0, 3);

        v8f acc0 = {}, acc1 = {}, acc2 = {}, acc3 = {};
#pragma unroll
        for (int kq = 0; kq < 128; kq += 4) {
            v2f a0 = *(const v2f*)(aBase + (kq + 0) * 64);
            v2f b0 = *(const v2f*)(bBase + (kq + 0) * 64);
            v2f a1 = *(const v2f*)(aBase + (kq + 1) * 64);
            v2f b1 = *(const v2f*)(bBase + (kq + 1) * 64);
            v2f a2 = *(const v2f*)(aBase + (kq + 2) * 64);
            v2f b2 = *(const v2f*)(bBase + (kq + 2) * 64);
            v2f a3 = *(const v2f*)(aBase + (kq + 3) * 64);
            v2f b3 = *(const v2f*)(bBase + (kq + 3) * 64);
            acc0 = __builtin_amdgcn_wmma_f32_16x16x4_f32(false, a0, false, b0, (short)0, acc0, false, false);
            acc1 = __builtin_amdgcn_wmma_f32_16x16x4_f32(false, a1, false, b1, (short)0, acc1, false, false);
            acc2 = __builtin_amdgcn_wmma_f32_16x16x4_f32(false, a2, false, b2, (short)0, acc2, false, false);
            acc3 = __builtin_amdgcn_wmma_f32_16x16x4_f32(false, a3, false, b3, (short)0, acc3, false, false);
        }
        v8f accv = (acc0 + acc1) + (acc2 + acc3);

        const int   nIdx = nt * 16 + (lane & 15);
        const float h    = hn[nIdx];
#pragma unroll
        for (int v = 0; v < 8; ++v) {
            float s = accv[v] - h;                // strict >: keeps lowest index on ties
            if (s > bval[v]) { bval[v] = s; bidx[v] = nIdx; }
        }
    }

    // Reduce across the 16 lanes holding the same row (xor 1,2,4,8 stays in group).
#pragma unroll
    for (int m = 1; m <= 8; m <<= 1) {
#pragma unroll
        for (int v = 0; v < 8; ++v) {
            float ov = __shfl_xor(bval[v], m, 32);
            int   oi = __shfl_xor(bidx[v], m, 32);
            if (ov > bval[v] || (ov == bval[v] && oi < bidx[v])) { bval[v] = ov; bidx[v] = oi; }
        }
    }
    if ((lane & 15) == 0) {   // lane 0 -> rows v, lane 16 -> rows v+8
        int rbase = m0 + wave * 16 + ((lane >> 4) << 3);
#pragma unroll
        for (int v = 0; v < 8; ++v) {
            best_score[rbase + v] = bval[v];
            best_idx[rbase + v]   = bidx[v];
        }
    }
}

// ---------------------------------------------------------------------------
// Per-row update: ||r||^2, residual -= codebook[idx], loss += ||r||^2-2*score,
// counts[idx]++, emit index as float into all_indices[(row)*Q + q].
// One wave32 per row, 8 waves per block.
// ---------------------------------------------------------------------------
__global__ void vq_update(float* __restrict__ residual, const float* __restrict__ cb,
                          const float* __restrict__ bscore, const int* __restrict__ bidx_,
                          float* __restrict__ counts, float* __restrict__ loss_accum,
                          float* __restrict__ out_idx, int q) {
    const int wave = threadIdx.x >> 5, lane = threadIdx.x & 31;
    const int row  = blockIdx.x * 8 + wave;
    const int idx  = bidx_[row];
    const float4* c = (const float4*)(cb + (size_t)idx * CDIM);
    float4*       r = (float4*)(residual + (size_t)row * CDIM);
    float r2 = 0.f;
#pragma unroll
    for (int j = 0; j < 4; ++j) {
        float4 rv = r[lane + 32 * j];
        float4 cv = c[lane + 32 * j];
        r2 += rv.x * rv.x + rv.y * rv.y + rv.z * rv.z + rv.w * rv.w;
        rv.x -= cv.x; rv.y -= cv.y; rv.z -= cv.z; rv.w -= cv.w;
        r[lane + 32 * j] = rv;
    }
#pragma unroll
    for (int m = 16; m >= 1; m >>= 1) r2 += __shfl_xor(r2, m, 32);
    if (lane == 0) {
        atomicAdd(loss_accum, r2 - 2.0f * bscore[row]);   // = min squared distance
        atomicAdd(&counts[idx], 1.0f);
        out_idx[(size_t)row * QSTAGES + q] = (float)idx;
    }
}

// ---------------------------------------------------------------------------
// Per-stage scalars: perplexity from counts, mean commit loss.
// ---------------------------------------------------------------------------
__global__ void vq_stage_final(const float* __restrict__ counts, const float* __restrict__ loss_accum,
                               float* __restrict__ stage_loss, float* __restrict__ stage_perp, int q) {
    __shared__ float red[256];
    float s = 0.f;
    for (int k = threadIdx.x; k < KCODES; k += 256) {
        float p = counts[k] * (1.0f / (32768.0f + 1e-10f));
        s += p * logf(p + 1e-7f);
    }
    red[threadIdx.x] = s;
    __syncthreads();
    for (int off = 128; off >= 1; off >>= 1) {
        if (threadIdx.x < off) red[threadIdx.x] += red[threadIdx.x + off];
        __syncthreads();
    }
    if (threadIdx.x == 0) {
        stage_perp[q] = expf(-red[0]);
        stage_loss[q] = loss_accum[0] * (1.0f / ((float)NT_ROWS * (float)CDIM));
    }
}

__global__ void vq_final_scalars(const float* __restrict__ stage_loss,
                                 const float* __restrict__ stage_perp,
                                 float* __restrict__ out_scalars) {
    if (threadIdx.x == 0) {
        float l = 0.f, p = 0.f;
        for (int q = 0; q < QSTAGES; ++q) { l += stage_loss[q]; p += stage_perp[q]; }
        out_scalars[0] = l * (1.0f / 6.0f);
        out_scalars[1] = p * (1.0f / 6.0f);
    }
}

// ---------------------------------------------------------------------------
extern "C" void kernel_launch(void* const* d_in, const int* in_sizes, int n_in,
                              void* d_out, int out_size, void* d_ws, size_t ws_size,
                              hipStream_t stream) {
    const float* x   = (const float*)d_in[0];   // (64, 512, 512)
    const float* cbs = (const float*)d_in[1];   // (6, 1024, 512)

    float* out_q   = (float*)d_out;                             // 64*512*512
    float* out_idx = out_q + (size_t)NB * CDIM * TDIM;          // 32768*6 (as float)
    float* out_sc  = out_idx + (size_t)NT_ROWS * QSTAGES;       // [loss, perp]

    char*  ws         = (char*)d_ws;
    float* residual   = (float*)(ws);                               // 64 MB
    float* best_score = (float*)(ws + 67108864);                    // 128 KB
    int*   best_idx   = (int*)  (ws + 67108864 + 131072);           // 128 KB
    float* hn         = (float*)(ws + 67108864 + 262144);           // 4 KB
    float* counts     = (float*)(ws + 67108864 + 262144 + 4096);    // 4 KB
    float* loss_accum = (float*)(ws + 67108864 + 262144 + 8192);
    float* stage_loss = (float*)(ws + 67108864 + 262144 + 8448);
    float* stage_perp = (float*)(ws + 67108864 + 262144 + 8704);

    dim3 tb(32, 8, 1);
    dim3 tg(TDIM / 32, CDIM / 32, NB);
    vq_transpose_in<<<tg, tb, 0, stream>>>(x, residual);

    for (int q = 0; q < QSTAGES; ++q) {
        const float* cb = cbs + (size_t)q * KCODES * CDIM;
        vq_stage_init<<<128, 256, 0, stream>>>(cb, hn, counts, loss_accum);
        vq_gemm_argmax<<<NT_ROWS / 64, 128, 0, stream>>>(residual, cb, hn, best_score, best_idx);
        vq_update<<<NT_ROWS / 8, 256, 0, stream>>>(residual, cb, best_score, best_idx,
                                                   counts, loss_accum, out_idx, q);
        vq_stage_final<<<1, 256, 0, stream>>>(counts, loss_accum, stage_loss, stage_perp, q);
    }
    vq_final_scalars<<<1, 64, 0, stream>>>(stage_loss, stage_perp, out_sc);
    vq_transpose_out<<<tg, tb, 0, stream>>>(x, residual, out_q);
}